// IsoMaxLossFirstPart_43241730736279
// MI455X (gfx1250) — compile-verified
//
#include <hip/hip_runtime.h>
#include <hip/hip_bf16.h>
#include <math.h>

typedef float v2f __attribute__((ext_vector_type(2)));
typedef float v4f __attribute__((ext_vector_type(4)));
typedef float v8f __attribute__((ext_vector_type(8)));

#define DK 128
#define DEPS 1e-6f

__device__ __forceinline__ float wave_reduce_sum(float v) {
    // wave32 butterfly reduction
    v += __shfl_xor(v, 16, 32);
    v += __shfl_xor(v, 8, 32);
    v += __shfl_xor(v, 4, 32);
    v += __shfl_xor(v, 2, 32);
    v += __shfl_xor(v, 1, 32);
    return v;
}

// One wave per row: lane handles 4 consecutive floats of a 128-float row.
__global__ void row_stats_kernel(const float* __restrict__ src, int validRows, int allocRows,
                                 float* __restrict__ norms, float* __restrict__ sums) {
    int wave = (int)((blockIdx.x * blockDim.x + threadIdx.x) >> 5);
    int lane = (int)(threadIdx.x & 31);
    if (wave >= allocRows) return;
    float nv = 0.0f, sv = 0.0f;
    if (wave < validRows) {
        v4f x = *(const v4f*)(src + (size_t)wave * DK + lane * 4);
        nv = x.x * x.x + x.y * x.y + x.z * x.z + x.w * x.w;
        sv = x.x + x.y + x.z + x.w;
    }
    nv = wave_reduce_sum(nv);
    sv = wave_reduce_sum(sv);
    if (lane == 0) { norms[wave] = nv; sums[wave] = sv; }
}

// 16x16 f32 dot-product tile via V_WMMA_F32_16X16X4_F32, K = 128.
// Computes A * B^T (both matrices row-major, K contiguous).
// Out-of-range rows are clamped to row 0 for address safety; the resulting
// garbage D rows/columns are masked in the caller's epilogue, so no fragment
// zeroing is needed in the inner loop (valid (m,n) outputs only consume
// valid A[m][*] and B[*][n] fragments).
__device__ __forceinline__ v8f wmma_tile_dot(const float* __restrict__ matA, int rowBase, int validA,
                                             const float* __restrict__ matB, int colBase, int validB,
                                             int lane) {
    int hi = lane >> 4;     // 0: K pair {0,1}; 1: K pair {2,3} within each x4 step
    int mn = lane & 15;
    int aRow = rowBase + mn;
    int bRow = colBase + mn;
    const float* ap = matA + (size_t)(aRow < validA ? aRow : 0) * DK + hi * 2;
    const float* bp = matB + (size_t)(bRow < validB ? bRow : 0) * DK + hi * 2;

    v8f acc = {0.0f, 0.0f, 0.0f, 0.0f, 0.0f, 0.0f, 0.0f, 0.0f};
    #pragma unroll
    for (int k0 = 0; k0 < DK; k0 += 4) {
        v2f a = *(const v2f*)(ap + k0);
        v2f b = *(const v2f*)(bp + k0);
        acc = __builtin_amdgcn_wmma_f32_16x16x4_f32(
            /*neg_a=*/false, a, /*neg_b=*/false, b,
            /*c_mod=*/(short)0, acc, /*reuse_a=*/false, /*reuse_b=*/false);
    }
    return acc;
}

// Per-tile partial sums of prototype-to-prototype distances (W vs W).
__global__ void proto_pairwise_kernel(const float* __restrict__ W,
                                      const float* __restrict__ nW, const float* __restrict__ Sw,
                                      int C, int CT, int numTiles,
                                      float* __restrict__ partials) {
    int t = (int)((blockIdx.x * blockDim.x + threadIdx.x) >> 5);
    int lane = (int)(threadIdx.x & 31);
    if (t >= numTiles) return;
    int tr = t / CT, tc = t % CT;

    v8f acc = wmma_tile_dot(W, tr * 16, C, W, tc * 16, C, lane);

    int hi = lane >> 4, mn = lane & 15;
    int col = tc * 16 + mn;
    float lsum = 0.0f;
    if (col < C) {
        float nc = nW[col], sc = Sw[col];
        #pragma unroll
        for (int v = 0; v < 8; ++v) {
            int row = tr * 16 + hi * 8 + v;        // C/D layout: VGPR v -> M = v + 8*hi
            if (row < C) {
                float d2 = nW[row] + nc + (float)DK * DEPS * DEPS
                         + 2.0f * DEPS * (Sw[row] - sc) - 2.0f * acc[v];
                lsum += sqrtf(fmaxf(d2, 0.0f));
            }
        }
    }
    lsum = wave_reduce_sum(lsum);
    if (lane == 0) partials[t] = lsum;
}

// Deterministic fixed-order reduction of tile partials -> scalar.
__global__ void reduce_partials_kernel(const float* __restrict__ partials, int n,
                                       float* __restrict__ outScalar) {
    __shared__ float sm[256];
    float s = 0.0f;
    for (int i = (int)threadIdx.x; i < n; i += 256) s += partials[i];
    sm[threadIdx.x] = s;
    __syncthreads();
    for (int st = 128; st > 0; st >>= 1) {
        if ((int)threadIdx.x < st) sm[threadIdx.x] += sm[threadIdx.x + st];
        __syncthreads();
    }
    if (threadIdx.x == 0) outScalar[0] = sm[0];
}

// Main: out[b,c] = protoSum - sqrt(||x_b - w_c + eps||^2), via X*W^T WMMA tiles.
__global__ void main_logits_kernel(const float* __restrict__ X, const float* __restrict__ W,
                                   const float* __restrict__ nX, const float* __restrict__ Sx,
                                   const float* __restrict__ nW, const float* __restrict__ Sw,
                                   const float* __restrict__ protoSumPtr,
                                   int Brows, int C, int CT, int numTiles,
                                   float* __restrict__ out) {
    int t = (int)((blockIdx.x * blockDim.x + threadIdx.x) >> 5);
    int lane = (int)(threadIdx.x & 31);
    if (t >= numTiles) return;
    int tr = t / CT, tc = t % CT;

    v8f acc = wmma_tile_dot(X, tr * 16, Brows, W, tc * 16, C, lane);

    float proto = protoSumPtr[0];
    int hi = lane >> 4, mn = lane & 15;
    int col = tc * 16 + mn;
    if (col < C) {
        float nc = nW[col], sc = Sw[col];
        #pragma unroll
        for (int v = 0; v < 8; ++v) {
            int row = tr * 16 + hi * 8 + v;
            if (row < Brows) {
                float d2 = nX[row] + nc + (float)DK * DEPS * DEPS
                         + 2.0f * DEPS * (Sx[row] - sc) - 2.0f * acc[v];
                out[(size_t)row * C + col] = proto - sqrtf(fmaxf(d2, 0.0f));
            }
        }
    }
}

extern "C" void kernel_launch(void* const* d_in, const int* in_sizes, int n_in,
                              void* d_out, int out_size, void* d_ws, size_t ws_size,
                              hipStream_t stream) {
    const float* X = (const float*)d_in[0];   // (B, 128)
    const float* W = (const float*)d_in[1];   // (C, 128)
    float* out = (float*)d_out;               // (B, C)

    const int Brows = in_sizes[0] / DK;       // 2048
    const int C     = in_sizes[1] / DK;       // 1000
    const int RT = (Brows + 15) / 16;         // 128 row tiles
    const int CT = (C + 15) / 16;             // 63 col tiles
    const int pcount = CT * CT;               // proto tiles

    // Workspace layout (floats): [0]=protoSum, [16..)=partials, then stats.
    float* ws       = (float*)d_ws;
    float* protoSum = ws;
    float* partials = ws + 16;
    float* nX = partials + pcount;
    float* Sx = nX + RT * 16;
    float* nW = Sx + RT * 16;
    float* Sw = nW + CT * 16;

    // 1) row stats (one wave per row, 8 waves per 256-thread block)
    row_stats_kernel<<<(RT * 16 + 7) / 8, 256, 0, stream>>>(X, Brows, RT * 16, nX, Sx);
    row_stats_kernel<<<(CT * 16 + 7) / 8, 256, 0, stream>>>(W, C, CT * 16, nW, Sw);

    // 2) prototype pairwise distance tile partials (W * W^T via WMMA)
    proto_pairwise_kernel<<<(pcount + 7) / 8, 256, 0, stream>>>(W, nW, Sw, C, CT, pcount, partials);

    // 3) deterministic scalar reduction
    reduce_partials_kernel<<<1, 256, 0, stream>>>(partials, pcount, protoSum);

    // 4) main GEMM + fused sqrt epilogue (X * W^T via WMMA)
    const int tiles = RT * CT;
    main_logits_kernel<<<(tiles + 7) / 8, 256, 0, stream>>>(
        X, W, nX, Sx, nW, Sw, protoSum, Brows, C, CT, tiles, out);
}